// LevyFieldSampler_63651415327274
// MI455X (gfx1250) — compile-verified
//
#include <hip/hip_runtime.h>
#include <stdint.h>

// ---------------------------------------------------------------------------
// Levy-flight token sampler for MI455X (gfx1250).
//
// Plan: 1 workgroup per batch row (4096 rows), 512 threads (16 wave32).
// Phase A: stream the 128000-float row from HBM ONCE using a 4-deep
//   GLOBAL_LOAD_ASYNC_TO_LDS pipeline (ASYNCcnt / s_wait_asynccnt), compute
//   e = exp(logit), park e as bf16 in LDS (250 KB row fits CDNA5's 320 KB
//   WGP LDS), and accumulate per-chunk (1024-elem) partial sums per wave.
// Phase B: wave 0 builds the 125-entry chunk-prefix CDF in LDS; then for
//   each of 8 samples do a two-level inverse-CDF search (chunk select +
//   block-wide scan over 1024 bf16 values) entirely from LDS, then apply
//   the CMS alpha-stable (alpha = phi) Levy step, sign, clip, write int32.
// HBM traffic = 2.1 GB read + 128 KB write  ->  ~90 us roofline.
// ---------------------------------------------------------------------------

#define VOCAB   128000
#define NS      8
#define CHUNK   1024
#define NCHUNK  125          // 125 * 1024 == 128000
#define BLOCK   512
#define NWAVE   (BLOCK / 32)
#define NSLOT   4            // async pipeline depth

// LDS layout (byte offsets inside dynamic shared)
#define OFF_EROW   0                              // bf16[128000] = 256000 B
#define OFF_STAGE  256000                         // f32 staging: 4 * 4 KB
#define OFF_PART   (OFF_STAGE + NSLOT*CHUNK*4)    // 272384: f32[16][125]
#define OFF_P      (OFF_PART + NWAVE*NCHUNK*4)    // 280384: f32[125] incl. prefix
#define OFF_WT     (OFF_P + NCHUNK*4)             // 280884: f32[16] wave totals
#define OFF_CTRL   (OFF_WT + NWAVE*4)             // 280948: int winner, int chunk
#define SMEM_BYTES (OFF_CTRL + 16)                // 280964 B  (< 320 KB WGP LDS)

// ---------------- small device helpers ----------------

__device__ __forceinline__ uint64_t splitmix64(uint64_t z) {
  z += 0x9E3779B97F4A7C15ull;
  z = (z ^ (z >> 30)) * 0xBF58476D1CE4E5B9ull;
  z = (z ^ (z >> 27)) * 0x94D049BB133111EBull;
  return z ^ (z >> 31);
}
// uniform in (0,1]; never 0, but top value rounds to 1.0f
__device__ __forceinline__ float u01(uint64_t h) {
  return ((float)((h >> 40) & 0xFFFFFFull) + 0.5f) * (1.0f / 16777216.0f);
}
__device__ __forceinline__ uint16_t f2bf(float x) {          // RNE f32->bf16
  uint32_t b = __float_as_uint(x);
  return (uint16_t)((b + 0x7FFFu + ((b >> 16) & 1u)) >> 16);
}
__device__ __forceinline__ float bf2f(uint32_t h) {
  return __uint_as_float(h << 16);
}

template <int N>
__device__ __forceinline__ void waitAsync() {
  // CDNA5 split counter: wait until <= N async LDS copies outstanding
  asm volatile("s_wait_asynccnt %0" :: "n"(N) : "memory");
}

// One wave32 instruction copies 32 lanes x 8 B = 256 B; 16 waves cover a
// full 4 KB chunk. Each lane copies exactly the bytes it later consumes,
// so completion needs only the wave-local ASYNCcnt (no block barrier).
__device__ __forceinline__ void issueChunk(char* smem, const float* rowp,
                                           int c, int tid) {
  const int slot = c & (NSLOT - 1);
  uint32_t lds = (uint32_t)(uintptr_t)(smem + OFF_STAGE + slot * CHUNK * 4 + tid * 8);
  const float* g = rowp + (size_t)c * CHUNK + tid * 2;
  asm volatile("global_load_async_to_lds_b64 %0, %1, off"
               :: "v"(lds), "v"(g) : "memory");
}

__device__ __forceinline__ void processChunk(char* smem, int c, int tid,
                                             int lane, int wid) {
  const int slot = c & (NSLOT - 1);
  const float2 x =
      *reinterpret_cast<const float2*>(smem + OFF_STAGE + slot * CHUNK * 4 + tid * 8);
  float e0 = __expf(x.x);
  float e1 = __expf(x.y);
  // park exp-values as packed bf16 pair in the resident row
  uint32_t pk = (uint32_t)f2bf(e0) | ((uint32_t)f2bf(e1) << 16);
  *reinterpret_cast<uint32_t*>(smem + (size_t)(c * CHUNK + tid * 2) * 2) = pk;
  // per-wave partial sum for this chunk (f32 accuracy, from pre-bf16 values)
  float p = e0 + e1;
  #pragma unroll
  for (int m = 16; m >= 1; m >>= 1) p += __shfl_xor(p, m, 32);
  if (lane == 0)
    reinterpret_cast<float*>(smem + OFF_PART)[wid * NCHUNK + c] = p;
}

// ---------------- kernel ----------------

__global__ __launch_bounds__(BLOCK)
void LevyFieldSampler_kernel(const float* __restrict__ logits,
                             int* __restrict__ out, int batch) {
  extern __shared__ char smem[];
  const int tid  = threadIdx.x;
  const int lane = tid & 31;
  const int wid  = tid >> 5;
  const int row  = blockIdx.x;
  if (row >= batch) return;

  const float* rowp = logits + (size_t)row * VOCAB;
  float* P       = reinterpret_cast<float*>(smem + OFF_P);
  float* waveTot = reinterpret_cast<float*>(smem + OFF_WT);
  int*   winner  = reinterpret_cast<int*>(smem + OFF_CTRL);
  int*   sChunk  = reinterpret_cast<int*>(smem + OFF_CTRL) + 1;

  // ---------------- Phase A: single HBM pass, async pipelined ----------------
  issueChunk(smem, rowp, 0, tid);
  issueChunk(smem, rowp, 1, tid);
  issueChunk(smem, rowp, 2, tid);
  issueChunk(smem, rowp, 3, tid);
  for (int c = 0; c <= NCHUNK - 5; ++c) {        // c = 0..120
    waitAsync<3>();                              // oldest (chunk c) landed in LDS
    processChunk(smem, c, tid, lane, wid);
    issueChunk(smem, rowp, c + 4, tid);          // refill the slot just drained
  }
  waitAsync<3>(); processChunk(smem, NCHUNK - 4, tid, lane, wid);
  waitAsync<2>(); processChunk(smem, NCHUNK - 3, tid, lane, wid);
  waitAsync<1>(); processChunk(smem, NCHUNK - 2, tid, lane, wid);
  waitAsync<0>(); processChunk(smem, NCHUNK - 1, tid, lane, wid);
  __syncthreads();

  // ---------------- chunk sums -> inclusive prefix CDF (wave 0) ----------------
  if (wid == 0) {
    const float* part = reinterpret_cast<const float*>(smem + OFF_PART);
    float sloc[4];
    float run = 0.f;
    #pragma unroll
    for (int j = 0; j < 4; ++j) {
      int c = lane * 4 + j;
      float cs = 0.f;
      if (c < NCHUNK) {
        #pragma unroll
        for (int w = 0; w < NWAVE; ++w) cs += part[w * NCHUNK + c];
      }
      run += cs;
      sloc[j] = run;                         // inclusive within lane's group
    }
    float tot = run, sc = run;
    #pragma unroll
    for (int off = 1; off < 32; off <<= 1) { // wave inclusive scan of lane totals
      float n = __shfl_up(sc, off, 32);
      if (lane >= off) sc += n;
    }
    float excl = sc - tot;
    #pragma unroll
    for (int j = 0; j < 4; ++j) {
      int c = lane * 4 + j;
      if (c < NCHUNK) P[c] = excl + sloc[j];
    }
  }
  __syncthreads();

  // ---------------- Phase B: 8 inverse-CDF draws, all from LDS ----------------
  const float total = P[NCHUNK - 1];
  for (int s = 0; s < NS; ++s) {
    __syncthreads();
    if (tid == 0) { *sChunk = NCHUNK - 1; *winner = CHUNK - 1; }
    __syncthreads();

    const uint64_t key = (uint64_t)row * NS + (uint64_t)s;
    const float T = u01(splitmix64(key ^ 0xC0FFEE1234567891ull)) * total;

    // level 1: which 1024-elem chunk
    if (tid < NCHUNK) {
      float lo = tid ? P[tid - 1] : 0.0f;
      float hi = P[tid];
      if (T >= lo && (T < hi || tid == NCHUNK - 1)) atomicMin(sChunk, tid);
    }
    __syncthreads();
    const int c = *sChunk;
    const float Tl = T - (c ? P[c - 1] : 0.0f);

    // level 2: block-wide ordered scan over the chunk's 1024 bf16 exp-values
    uint32_t pk = *reinterpret_cast<const uint32_t*>(
        smem + (size_t)(c * CHUNK + tid * 2) * 2);
    float e0 = bf2f(pk & 0xFFFFu);
    float e1 = bf2f(pk >> 16);
    float p = e0 + e1, sc = p;
    #pragma unroll
    for (int off = 1; off < 32; off <<= 1) {
      float n = __shfl_up(sc, off, 32);
      if (lane >= off) sc += n;
    }
    if (lane == 31) waveTot[wid] = sc;
    __syncthreads();
    float woff = 0.f;
    for (int w = 0; w < wid; ++w) woff += waveTot[w];
    float ex = woff + sc - p;                  // exclusive prefix before e0
    float c0 = ex + e0, c1 = c0 + e1;
    if (Tl >= ex && Tl < c0) atomicMin(winner, tid * 2);
    if (Tl >= c0 && Tl < c1) atomicMin(winner, tid * 2 + 1);
    __syncthreads();

    if (tid == 0) {
      int cur = c * CHUNK + *winner;           // categorical sample

      // ---- CMS symmetric alpha-stable step, alpha = phi, scale = 0.1 ----
      const float A = 1.61803398874989f;
      float r1 = u01(splitmix64(key ^ 0xA5A5A5A5DEADBEEFull));
      float r2 = u01(splitmix64(key ^ 0x0123456789ABCDEFull));
      uint64_t hd = splitmix64(key ^ 0x0F0F0F0F0F0F0F0Full);
      float uu = (r1 - 0.5f) * 3.14159265358979f;     // (-pi/2, pi/2)
      float w  = -__logf(r2);                          // Exp(1)
      float t1 = __sinf(A * uu) * __powf(__cosf(uu), -1.0f / A);
      float t2 = __powf(__cosf((1.0f - A) * uu) / w, (1.0f - A) / A);
      float levy = 0.1f * t1 * t2;
      float step = fabsf(levy) * (0.1f * (float)VOCAB);
      float dlt  = (hd & 1ull) ? step : -step;         // sign(normal) == fair coin
      dlt = fminf(fmaxf(dlt, -2.0e9f), 2.0e9f);        // keep int cast defined
      long long np = (long long)cur + (long long)(int)dlt;  // trunc toward zero
      if (np < 0) np = 0;
      if (np > VOCAB - 1) np = VOCAB - 1;
      out[(size_t)row * NS + s] = (int)np;             // (batch, num_samples)
    }
  }
}

// ---------------- launcher ----------------

extern "C" void kernel_launch(void* const* d_in, const int* in_sizes, int n_in,
                              void* d_out, int out_size, void* d_ws, size_t ws_size,
                              hipStream_t stream) {
  (void)n_in; (void)d_ws; (void)ws_size; (void)out_size;
  const float* logits = (const float*)d_in[0];
  // d_in[1] = num_samples (device scalar, == 8); compile-time NS matches it.
  const int batch = in_sizes[0] / VOCAB;
  int* out = (int*)d_out;   // reference output dtype is int32
  dim3 grid(batch), block(BLOCK);
  LevyFieldSampler_kernel<<<grid, block, SMEM_BYTES, stream>>>(logits, out, batch);
}